// ComputeTargets_48996986913429
// MI455X (gfx1250) — compile-verified
//
#include <hip/hip_runtime.h>
#include <stdint.h>

#define NUM_CLASSES 80
#define NBOX 64

typedef float v4f __attribute__((ext_vector_type(4)));
typedef int v4i __attribute__((ext_vector_type(4)));

#if __has_builtin(__builtin_amdgcn_global_load_async_to_lds_b128) && \
    __has_builtin(__builtin_amdgcn_s_wait_asynccnt)
#define USE_ASYNC_LDS 1
typedef __attribute__((address_space(1))) v4i as1_v4i;
typedef __attribute__((address_space(3))) v4i as3_v4i;
#endif

// One block = 256 threads = 8 wave32, one anchor per thread, one batch per blockIdx.y.
// Boxes (64 x 5 floats = 1280 B) are staged in LDS via async global->LDS DMA.
// cls one-hot rows are written cooperatively per wave: 32 anchors x 80 floats
// = 640 float4s, 20 fully-coalesced 128-bit NT stores per lane.
__global__ __launch_bounds__(256) void ComputeTargets_48996986913429_kernel(
    const float* __restrict__ ann, const float* __restrict__ anchors,
    float* __restrict__ out, int A, int B) {
#pragma clang fp contract(off)
  __shared__ __align__(16) float sbox[NBOX * 5];
  const int tid = threadIdx.x;
  const int b = blockIdx.y;
  const float* annb = ann + (size_t)b * (NBOX * 5);

#ifdef USE_ASYNC_LDS
  if (tid < (NBOX * 5) / 4) {  // 80 lanes x 16 B = 1280 B
    __builtin_amdgcn_global_load_async_to_lds_b128(
        (as1_v4i*)(annb + tid * 4), (as3_v4i*)(&sbox[tid * 4]), 0, 0);
  }
  __builtin_amdgcn_s_wait_asynccnt(0);
#else
  for (int i = tid; i < NBOX * 5; i += blockDim.x) sbox[i] = annb[i];
#endif
  __syncthreads();

  const int a = blockIdx.x * 256 + tid;
  const int waveBase = a & ~31;
  if (waveBase >= A) return;  // A % 32 == 0 -> each wave fully in or out of range

  const v4f anc = ((const v4f*)anchors)[a];  // x0,y0,x1,y1
  const float area_a = (anc.z - anc.x) * (anc.w - anc.y);

  // First-max argmax over 64 boxes (matches jnp.argmax tie semantics via strict >).
  float best_iou = -1.0f;
  int best_i = 0;
  for (int j = 0; j < NBOX; ++j) {
    const float bx0 = sbox[j * 5 + 0], by0 = sbox[j * 5 + 1];
    const float bx1 = sbox[j * 5 + 2], by1 = sbox[j * 5 + 3];
    const float ltx = fmaxf(anc.x, bx0), lty = fmaxf(anc.y, by0);
    const float rbx = fminf(anc.z, bx1), rby = fminf(anc.w, by1);
    const float w = fmaxf(rbx - ltx, 0.0f);
    const float h = fmaxf(rby - lty, 0.0f);
    const float inter = w * h;
    const float area_b = (bx1 - bx0) * (by1 - by0);
    const float uni = (area_a + area_b) - inter;
    const float iou = inter / fmaxf(uni, 1e-8f);
    if (iou > best_iou) { best_iou = iou; best_i = j; }
  }

  const bool positive = best_iou >= 0.5f;
  const bool ignore = (best_iou > 0.4f) && !positive;
  const float state = positive ? 1.0f : (ignore ? -1.0f : 0.0f);

  const float wa = anc.z - anc.x;
  const float ha = anc.w - anc.y;
  v4f reg;
  reg.x = ((sbox[best_i * 5 + 0] - anc.x) / wa) / 0.2f;
  reg.y = ((sbox[best_i * 5 + 1] - anc.y) / ha) / 0.2f;
  reg.z = ((sbox[best_i * 5 + 2] - anc.z) / wa) / 0.2f;
  reg.w = ((sbox[best_i * 5 + 3] - anc.w) / ha) / 0.2f;

  const size_t clsSize = (size_t)B * (size_t)A * NUM_CLASSES;
  const size_t regSize = (size_t)B * (size_t)A * 4;
  const size_t ba = (size_t)b * (size_t)A + (size_t)a;
  __builtin_nontemporal_store(reg, ((v4f*)(out + clsSize)) + ba);
  __builtin_nontemporal_store(state, out + clsSize + regSize + ba);

  // Cooperative wave32 one-hot write: contiguous 32*80 floats starting at waveBase.
  const int v = positive ? (int)sbox[best_i * 5 + 4] : -1;
  const int lane = tid & 31;
  v4f* cls4 = (v4f*)(out + ((size_t)b * (size_t)A + (size_t)waveBase) * NUM_CLASSES);
  for (int it = 0; it < (NUM_CLASSES / 4); ++it) {  // 20 iterations
    const int f = it * 32 + lane;                 // float4 index within 640
    const int aL = f / (NUM_CLASSES / 4);         // local anchor 0..31
    const int c0 = (f % (NUM_CLASSES / 4)) * 4;   // class base 0..76
    const int vv = __shfl(v, aL, 32);             // ds_bpermute within wave32
    v4f r;
    r.x = (vv == c0) ? 1.0f : 0.0f;
    r.y = (vv == c0 + 1) ? 1.0f : 0.0f;
    r.z = (vv == c0 + 2) ? 1.0f : 0.0f;
    r.w = (vv == c0 + 3) ? 1.0f : 0.0f;
    __builtin_nontemporal_store(r, cls4 + f);
  }
}

extern "C" void kernel_launch(void* const* d_in, const int* in_sizes, int n_in,
                              void* d_out, int out_size, void* d_ws, size_t ws_size,
                              hipStream_t stream) {
  const float* ann = (const float*)d_in[0];      // (B, 64, 5) float32
  const float* anchors = (const float*)d_in[1];  // (A, 4) float32
  float* out = (float*)d_out;                    // cls (B,A,80) | reg (B,A,4) | states (B,A)
  const int A = in_sizes[1] / 4;                 // 120000
  const int B = in_sizes[0] / (NBOX * 5);        // 8
  dim3 grid((A + 255) / 256, B, 1);
  ComputeTargets_48996986913429_kernel<<<grid, dim3(256, 1, 1), 0, stream>>>(
      ann, anchors, out, A, B);
}